// ExLRestSelfAtten_26371099198089
// MI455X (gfx1250) — compile-verified
//
#include <hip/hip_runtime.h>

// ExLRestSelfAtten for MI455X (gfx1250, wave32, WMMA).
//
// Key reduction: softmax in the reference is over a singleton axis, so the
// attention weights are exactly 1.0. Hence:
//   out = fc3(relu(fc2( windowsum_33( relu(x@W1+b1) @ Wv ) ))),  a = ones.
// Wq/Wk/q/k are dead. Gather commutes with right-matmul, so V is computed
// once per position and the 33-neighborhood is a sliding sum.
//
// All GEMMs use V_WMMA_F32_16X16X4_F32 (fp32 in/out, matches fp32 reference).

typedef __attribute__((ext_vector_type(2))) float v2f;
typedef __attribute__((ext_vector_type(8))) float v8f;

#define DMODEL 128
#define SEQL   4096
#define NBATCH 4
#define NROWS  (NBATCH * SEQL)   // 16384
#define WIN    16                // half window; full = 33
#define HID    24
#define OUTD   2

__device__ __forceinline__ v8f wmma4(v2f a, v2f b, v8f c) {
    // 8-arg form: (neg_a, A, neg_b, B, c_mod, C, reuse_a, reuse_b)
    return __builtin_amdgcn_wmma_f32_16x16x4_f32(false, a, false, b,
                                                 (short)0, c, false, false);
}

// ---------------------------------------------------------------------------
// Kernel 1: V = relu(x @ W1 + b1) @ Wv      (two chained 16x128x128 GEMM tiles)
// One wave per 16-row tile; 4 waves / block.
// ---------------------------------------------------------------------------
__global__ __launch_bounds__(128) void k1_fc1_wv(
    const float* __restrict__ x, const float* __restrict__ w1,
    const float* __restrict__ b1, const float* __restrict__ wv,
    float* __restrict__ V)
{
    __shared__ float ldsH[4][16 * DMODEL];   // per-wave H tile (16x128)

    const int wave = threadIdx.x >> 5;
    const int lane = threadIdx.x & 31;
    const int half = lane >> 4;              // 0 or 1 (lane group)
    const int mod  = lane & 15;
    const int r0   = (blockIdx.x * 4 + wave) * 16;

    // ---- GEMM1: H = relu(X_tile @ W1 + b1) ----
    for (int nt = 0; nt < 8; ++nt) {
        const int col = nt * 16 + mod;
        v8f acc;
        const float bias = b1[col];          // C column depends only on lane
        #pragma unroll
        for (int i = 0; i < 8; ++i) acc[i] = bias;
        #pragma unroll 4
        for (int ks = 0; ks < 32; ++ks) {
            const int k = ks * 4 + half * 2;
            v2f a, b;
            a.x = x[(r0 + mod) * DMODEL + k];        // A[row=mod][k]
            a.y = x[(r0 + mod) * DMODEL + k + 1];
            b.x = w1[k * DMODEL + col];              // B[k][col]
            b.y = w1[(k + 1) * DMODEL + col];
            acc = wmma4(a, b, acc);
        }
        #pragma unroll
        for (int i = 0; i < 8; ++i) {                // D row = i + 8*half
            float h = acc[i] > 0.f ? acc[i] : 0.f;   // relu
            ldsH[wave][(half * 8 + i) * DMODEL + col] = h;
        }
    }
    __syncthreads();

    // ---- GEMM2: V_tile = H_tile @ Wv ----
    for (int nt = 0; nt < 8; ++nt) {
        const int col = nt * 16 + mod;
        v8f acc = {};
        #pragma unroll 4
        for (int ks = 0; ks < 32; ++ks) {
            const int k = ks * 4 + half * 2;
            v2f a, b;
            a.x = ldsH[wave][mod * DMODEL + k];
            a.y = ldsH[wave][mod * DMODEL + k + 1];
            b.x = wv[k * DMODEL + col];
            b.y = wv[(k + 1) * DMODEL + col];
            acc = wmma4(a, b, acc);
        }
        #pragma unroll
        for (int i = 0; i < 8; ++i)
            V[(r0 + half * 8 + i) * DMODEL + col] = acc[i];
    }
}

// ---------------------------------------------------------------------------
// Kernel 2: S = windowsum_33(V) ; T = relu(S @ fc2 + b2) (WMMA, N padded to 32)
//           out = T[:, :24] @ fc3 + b3
// ---------------------------------------------------------------------------
__global__ __launch_bounds__(128) void k2_win_fc23(
    const float* __restrict__ V, const float* __restrict__ w2,
    const float* __restrict__ b2, const float* __restrict__ w3,
    const float* __restrict__ b3, float* __restrict__ out)
{
    __shared__ float ldsS[4][16 * DMODEL];   // per-wave S tile (16x128)
    __shared__ float ldsW2[DMODEL * 32];     // fc2 zero-padded 128x32
    __shared__ float ldsT[4][16 * 32];       // per-wave fc2 output tile

    const int tid  = threadIdx.x;
    const int wave = tid >> 5;
    const int lane = tid & 31;
    const int half = lane >> 4;
    const int mod  = lane & 15;
    const int r0   = (blockIdx.x * 4 + wave) * 16;

    // stage fc2 weights, zero-padded N: 24 -> 32
    for (int idx = tid; idx < DMODEL * 32; idx += 128) {
        const int k = idx >> 5, n = idx & 31;
        ldsW2[idx] = (n < HID) ? w2[k * HID + n] : 0.f;
    }

    // sliding 33-window sum per batch (no cross-batch leakage).
    const int lo = (r0 >> 12) << 12;   // batch start (L = 4096)
    const int hi = lo + SEQL;
    for (int cc = 0; cc < 4; ++cc) {
        const int c = lane + 32 * cc;
        float s = 0.f;
        for (int j = r0 - WIN; j <= r0 + WIN; ++j)
            if (j >= lo && j < hi) s += V[j * DMODEL + c];
        ldsS[wave][c] = s;
        for (int row = 1; row < 16; ++row) {
            const int jadd = r0 + row + WIN;
            const int jsub = r0 + row - WIN - 1;
            if (jadd < hi)  s += V[jadd * DMODEL + c];
            if (jsub >= lo) s -= V[jsub * DMODEL + c];
            ldsS[wave][row * DMODEL + c] = s;
        }
    }
    __syncthreads();

    // fc2 via WMMA: T = relu(S @ W2pad + b2), 16x32 (cols 24..31 are zero)
    for (int nt = 0; nt < 2; ++nt) {
        const int col = nt * 16 + mod;
        v8f acc;
        const float bias = (col < HID) ? b2[col] : 0.f;
        #pragma unroll
        for (int i = 0; i < 8; ++i) acc[i] = bias;
        #pragma unroll 4
        for (int ks = 0; ks < 32; ++ks) {
            const int k = ks * 4 + half * 2;
            v2f a, b;
            a.x = ldsS[wave][mod * DMODEL + k];
            a.y = ldsS[wave][mod * DMODEL + k + 1];
            b.x = ldsW2[k * 32 + col];
            b.y = ldsW2[(k + 1) * 32 + col];
            acc = wmma4(a, b, acc);
        }
        #pragma unroll
        for (int i = 0; i < 8; ++i) {
            float t = acc[i] > 0.f ? acc[i] : 0.f;
            ldsT[wave][(half * 8 + i) * 32 + col] = t;
        }
    }
    __syncthreads();

    // fc3: one (row, o) pair per lane; 16 rows x 2 outputs = 32 lanes.
    {
        const int row = lane >> 1;
        const int o   = lane & 1;
        float s = b3[o];
        #pragma unroll
        for (int j = 0; j < HID; ++j)
            s += ldsT[wave][row * 32 + j] * w3[j * OUTD + o];
        out[(r0 + row) * OUTD + o] = s;
    }
}

// ---------------------------------------------------------------------------
// Kernel 3: attention map `a` — softmax over a singleton axis — exactly 1.0.
// ---------------------------------------------------------------------------
__global__ void k3_ones(float* __restrict__ a, int n) {
    const int i = blockIdx.x * blockDim.x + threadIdx.x;
    if (i < n) a[i] = 1.0f;
}

extern "C" void kernel_launch(void* const* d_in, const int* in_sizes, int n_in,
                              void* d_out, int out_size, void* d_ws, size_t ws_size,
                              hipStream_t stream)
{
    const float* x  = (const float*)d_in[0];
    const float* w1 = (const float*)d_in[1];
    const float* b1 = (const float*)d_in[2];
    // d_in[3] = Wq, d_in[4] = Wk  -> dead (softmax over singleton axis)
    const float* wv = (const float*)d_in[5];
    const float* w2 = (const float*)d_in[6];
    const float* b2 = (const float*)d_in[7];
    const float* w3 = (const float*)d_in[8];
    const float* b3 = (const float*)d_in[9];

    float* out  = (float*)d_out;                 // (4,4096,2)   = 32768 f32
    float* aout = out + (size_t)NROWS * OUTD;    // (4,4096,1,33)= 540672 f32
    float* V    = (float*)d_ws;                  // 16384x128 f32 = 8 MB scratch

    k1_fc1_wv  <<<NROWS / 64, 128, 0, stream>>>(x, w1, b1, wv, V);
    k2_win_fc23<<<NROWS / 64, 128, 0, stream>>>(V, w2, b2, w3, b3, out);

    const int na = NROWS * (2 * WIN + 1);
    k3_ones<<<(na + 255) / 256, 256, 0, stream>>>(aout, na);
}